// LSHAttention_44848048505358
// MI455X (gfx1250) — compile-verified
//
#include <hip/hip_runtime.h>
#include <hip/hip_bf16.h>
#include <math.h>

typedef __attribute__((ext_vector_type(2))) float v2f;
typedef __attribute__((ext_vector_type(8))) float v8f;

#define B_    8
#define S_    4096
#define D_    128
#define H_    8
#define NB_   64            // n_buckets == n_bins
#define CH_   64            // chunk size
#define C_    (H_ * NB_)    // 512 chunks per batch
#define HS_   (H_ * S_)     // 32768
#define SCALE_ 0.08838834764831845f   // 1/sqrt(128)
#define MASKV_ (-1e5f)

// ---------------------------------------------------------------------------
// 1) Hashing: rot = qk @ R  ([B,S,128] x [128,256]) ; bucket = argmax [rot,-rot]
// ---------------------------------------------------------------------------
__global__ __launch_bounds__(256) void hash_kernel(const float* __restrict__ qk,
                                                   const float* __restrict__ rotR,
                                                   int* __restrict__ bucket_i,
                                                   float* __restrict__ out_buckets) {
    __shared__ float qt[16 * 128];
    __shared__ float rot[16 * 256];
    int bid  = blockIdx.x;
    int b    = bid >> 8;            // S/16 = 256 tiles per batch
    int tile = bid & 255;
    int t0   = tile * 16;
    int tid  = threadIdx.x;

    for (int i = tid; i < 16 * 128; i += 256) {
        int r = i >> 7, cc = i & 127;
        qt[i] = qk[((size_t)(b * S_ + t0 + r)) * D_ + cc];
    }
    __syncthreads();

    {   // each thread owns one rotation column n (0..255)
        int n = tid;
        float acc[16];
#pragma unroll
        for (int t = 0; t < 16; ++t) acc[t] = 0.f;
        for (int k = 0; k < 128; ++k) {
            float rv = rotR[k * 256 + n];
#pragma unroll
            for (int t = 0; t < 16; ++t) acc[t] = fmaf(qt[t * 128 + k], rv, acc[t]);
        }
#pragma unroll
        for (int t = 0; t < 16; ++t) rot[t * 256 + n] = acc[t];
    }
    __syncthreads();

    if (tid < 128) {
        int t = tid >> 3, h = tid & 7;
        const float* rp = &rot[t * 256 + h * 32];
        float vmax = rp[0]; int imax = 0;
        float vmin = rp[0]; int imin = 0;
        for (int i = 1; i < 32; ++i) {
            float x = rp[i];
            if (x > vmax) { vmax = x; imax = i; }
            if (x < vmin) { vmin = x; imin = i; }
        }
        int bkt = (vmax >= -vmin) ? imax : (imin + 32);   // concat order [rot,-rot]
        int tok = t0 + t;
        bucket_i[((size_t)(b * H_ + h)) * S_ + tok] = bkt;
        out_buckets[(size_t)b * HS_ + h * S_ + tok] = (float)(bkt + h * NB_);
    }
}

// ---------------------------------------------------------------------------
// 2) Stable counting sort per (b,h): 64 buckets, 4096 tokens
// ---------------------------------------------------------------------------
__global__ __launch_bounds__(256) void sort_kernel(const int* __restrict__ bucket_i,
                                                   int* __restrict__ sorted_t,
                                                   int* __restrict__ posm) {
    __shared__ unsigned short hist[256 * 64];
    __shared__ int baseArr[64];
    int bh  = blockIdx.x;           // b*H + h
    int h   = bh & 7;
    int tid = threadIdx.x;
    const int* bp = bucket_i + (size_t)bh * S_;

#pragma unroll
    for (int j = 0; j < 64; ++j) hist[tid * 64 + j] = 0;
    __syncthreads();

    int t0 = tid * 16;
    for (int i = 0; i < 16; ++i) {
        int bk = bp[t0 + i];
        hist[tid * 64 + bk] = (unsigned short)(hist[tid * 64 + bk] + 1);
    }
    __syncthreads();

    if (tid < 64) {                 // exclusive prefix over thread slices, per bucket
        unsigned int run = 0;
        for (int i = 0; i < 256; ++i) {
            unsigned int tmp = hist[i * 64 + tid];
            hist[i * 64 + tid] = (unsigned short)run;
            run += tmp;
        }
        baseArr[tid] = (int)run;    // bucket total
    }
    __syncthreads();
    if (tid == 0) {                 // exclusive scan of bucket totals
        int run = 0;
        for (int i = 0; i < 64; ++i) { int tmp = baseArr[i]; baseArr[i] = run; run += tmp; }
    }
    __syncthreads();

    int* st = sorted_t + (size_t)bh * S_;
    int* pm = posm     + (size_t)bh * S_;
    for (int i = 0; i < 16; ++i) {
        int t  = t0 + i;
        int bk = bp[t];
        int r  = hist[tid * 64 + bk];
        hist[tid * 64 + bk] = (unsigned short)(r + 1);
        int j  = baseArr[bk] + r;
        st[j] = t;                  // slot -> token
        pm[t] = h * S_ + j;         // token -> global sorted slot (within batch)
    }
}

// ---------------------------------------------------------------------------
// 3) Per-chunk attention, pass 1: row LSE via fp32 WMMA (k-outer, 8 tiles live)
// ---------------------------------------------------------------------------
__global__ __launch_bounds__(128) void attn_lse_kernel(const float* __restrict__ qk,
                                                       const int* __restrict__ sorted_t,
                                                       float* __restrict__ lse) {
    __shared__ float qs[64 * 128];
    __shared__ float ks[128 * 128];
    __shared__ int   bqt[64];
    __shared__ int   bkt[128];

    int blk = blockIdx.x;
    int b   = blk >> 9;             // C_ = 512
    int c   = blk & (C_ - 1);
    int cp  = (c + C_ - 1) & (C_ - 1);
    int tid = threadIdx.x;
    const int* stb = sorted_t + (size_t)b * HS_;

    if (tid < 64) {
        int t = stb[c * CH_ + tid];
        bqt[tid] = t;
        bkt[tid] = t;
    } else {
        bkt[tid] = stb[cp * CH_ + (tid - 64)];
    }
    __syncthreads();

    for (int i = tid; i < 64 * 128; i += 128) {
        int r = i >> 7, col = i & 127;
        qs[i] = qk[((size_t)(b * S_) + bqt[r]) * D_ + col];
    }
    for (int i = tid; i < 128 * 128; i += 128) {
        int r = i >> 7, col = i & 127;
        ks[i] = qk[((size_t)(b * S_) + bkt[r]) * D_ + col];
    }
    __syncthreads();

    {   // normalize key rows: bk = k * rsqrt(sum(k*k) + eps)
        int r = tid;
        float s = 0.f;
        for (int e = 0; e < 128; ++e) { float x = ks[r * 128 + e]; s = fmaf(x, x, s); }
        float inv = 1.0f / sqrtf(s + 1e-6f);
        for (int e = 0; e < 128; ++e) ks[r * 128 + e] *= inv;
    }
    __syncthreads();

    int wave = tid >> 5;
    int lane = tid & 31;
    int m0   = wave * 16;
    int half = lane >> 4;           // 0: K=0,1 ; 1: K=2,3
    int l15  = lane & 15;
    int kb   = half * 2;

    v8f accs[8];
    {
        v8f z = {};
#pragma unroll
        for (int nt = 0; nt < 8; ++nt) accs[nt] = z;
    }
    const float* qrow = &qs[(m0 + l15) * 128];
#pragma unroll
    for (int k0 = 0; k0 < 128; k0 += 4) {
        v2f a;
        a.x = qrow[k0 + kb];
        a.y = qrow[k0 + kb + 1];
#pragma unroll
        for (int nt = 0; nt < 8; ++nt) {
            v2f bb;
            bb.x = ks[(nt * 16 + l15) * 128 + k0 + kb];
            bb.y = ks[(nt * 16 + l15) * 128 + k0 + kb + 1];
            accs[nt] = __builtin_amdgcn_wmma_f32_16x16x4_f32(false, a, false, bb,
                                                             (short)0, accs[nt], false, false);
        }
    }

    // scale + self-mask in registers
    float rm[8], rs[8];
#pragma unroll
    for (int r = 0; r < 8; ++r) rm[r] = -1e30f;
#pragma unroll
    for (int nt = 0; nt < 8; ++nt) {
        int kt = bkt[nt * 16 + l15];
#pragma unroll
        for (int r = 0; r < 8; ++r) {
            int rowc = m0 + r + half * 8;
            float dv = accs[nt][r] * SCALE_;
            if (bqt[rowc] == kt) dv = MASKV_;
            accs[nt][r] = dv;
            rm[r] = fmaxf(rm[r], dv);
        }
    }
#pragma unroll
    for (int r = 0; r < 8; ++r)
        for (int off = 1; off < 16; off <<= 1)
            rm[r] = fmaxf(rm[r], __shfl_xor(rm[r], off, 32));
#pragma unroll
    for (int r = 0; r < 8; ++r) rs[r] = 0.f;
#pragma unroll
    for (int nt = 0; nt < 8; ++nt)
#pragma unroll
        for (int r = 0; r < 8; ++r)
            rs[r] += expf(accs[nt][r] - rm[r]);
#pragma unroll
    for (int r = 0; r < 8; ++r)
        for (int off = 1; off < 16; off <<= 1)
            rs[r] += __shfl_xor(rs[r], off, 32);

    if (l15 == 0) {
#pragma unroll
        for (int r = 0; r < 8; ++r) {
            int rowc = m0 + r + half * 8;
            lse[(size_t)b * HS_ + c * CH_ + rowc] = rm[r] + logf(rs[r]);
        }
    }
}

// ---------------------------------------------------------------------------
// 4) Combine weights across hash rounds: w_h = exp(lse_h - LSE(all h))
// ---------------------------------------------------------------------------
__global__ __launch_bounds__(256) void combine_kernel(const int* __restrict__ posm,
                                                      const float* __restrict__ lse,
                                                      float* __restrict__ wgt) {
    int idx = blockIdx.x * 256 + threadIdx.x;
    if (idx >= B_ * S_) return;
    int b = idx / S_;
    int t = idx - b * S_;
    float l[8]; float m = -1e30f;
#pragma unroll
    for (int h = 0; h < 8; ++h) {
        int g = posm[((size_t)(b * H_ + h)) * S_ + t];
        l[h] = lse[(size_t)b * HS_ + g];
        m = fmaxf(m, l[h]);
    }
    float s = 0.f;
#pragma unroll
    for (int h = 0; h < 8; ++h) s += expf(l[h] - m);
    float L = m + logf(s);
#pragma unroll
    for (int h = 0; h < 8; ++h)
        wgt[((size_t)(b * H_ + h)) * S_ + t] = expf(l[h] - L);
}

__global__ __launch_bounds__(256) void zero_kernel(float* __restrict__ p, int n) {
    int i = blockIdx.x * 256 + threadIdx.x;
    if (i < n) p[i] = 0.f;
}

// ---------------------------------------------------------------------------
// 5) Per-chunk attention, pass 2: probs@V weighted by w, scatter-add to out
// ---------------------------------------------------------------------------
__global__ __launch_bounds__(128) void attn_out_kernel(const float* __restrict__ qk,
                                                       const float* __restrict__ v,
                                                       const int* __restrict__ sorted_t,
                                                       const float* __restrict__ lse,
                                                       const float* __restrict__ wgt,
                                                       float* __restrict__ out) {
    __shared__ float qs[64 * 128];          // reused as probs after dots
    __shared__ float ks[128 * 128];
    __shared__ float vs[128 * 128];
    __shared__ int   bqt[64];
    __shared__ int   bkt[128];
    __shared__ float lrow[64];
    __shared__ float wrow[64];

    int blk = blockIdx.x;
    int b   = blk >> 9;
    int c   = blk & (C_ - 1);
    int cp  = (c + C_ - 1) & (C_ - 1);
    int h   = c >> 6;                        // c / n_bins
    int tid = threadIdx.x;
    const int* stb = sorted_t + (size_t)b * HS_;

    if (tid < 64) {
        int t = stb[c * CH_ + tid];
        bqt[tid]  = t;
        bkt[tid]  = t;
        lrow[tid] = lse[(size_t)b * HS_ + c * CH_ + tid];
        wrow[tid] = wgt[((size_t)(b * H_ + h)) * S_ + t];
    } else {
        bkt[tid] = stb[cp * CH_ + (tid - 64)];
    }
    __syncthreads();

    for (int i = tid; i < 64 * 128; i += 128) {
        int r = i >> 7, col = i & 127;
        qs[i] = qk[((size_t)(b * S_) + bqt[r]) * D_ + col];
    }
    for (int i = tid; i < 128 * 128; i += 128) {
        int r = i >> 7, col = i & 127;
        int t = bkt[r];
        ks[i] = qk[((size_t)(b * S_) + t) * D_ + col];
        vs[i] = v [((size_t)(b * S_) + t) * D_ + col];
    }
    __syncthreads();

    {   // normalize key rows
        int r = tid;
        float s = 0.f;
        for (int e = 0; e < 128; ++e) { float x = ks[r * 128 + e]; s = fmaf(x, x, s); }
        float inv = 1.0f / sqrtf(s + 1e-6f);
        for (int e = 0; e < 128; ++e) ks[r * 128 + e] *= inv;
    }
    __syncthreads();

    int wave = tid >> 5;
    int lane = tid & 31;
    int m0   = wave * 16;
    int half = lane >> 4;
    int l15  = lane & 15;
    int kb   = half * 2;

    // dots: k-outer, all 8 N-tiles live; A loaded once per k-step
    v8f accs[8];
    {
        v8f z = {};
#pragma unroll
        for (int nt = 0; nt < 8; ++nt) accs[nt] = z;
    }
    {
        const float* qrow = &qs[(m0 + l15) * 128];
#pragma unroll
        for (int k0 = 0; k0 < 128; k0 += 4) {
            v2f a;
            a.x = qrow[k0 + kb];
            a.y = qrow[k0 + kb + 1];
#pragma unroll
            for (int nt = 0; nt < 8; ++nt) {
                v2f bb;
                bb.x = ks[(nt * 16 + l15) * 128 + k0 + kb];
                bb.y = ks[(nt * 16 + l15) * 128 + k0 + kb + 1];
                accs[nt] = __builtin_amdgcn_wmma_f32_16x16x4_f32(false, a, false, bb,
                                                                 (short)0, accs[nt], false, false);
            }
        }
    }

    // probs = exp(scaled_masked_dot - lse_row) * w_row  -> into qs region
    float* ps = qs;
#pragma unroll
    for (int nt = 0; nt < 8; ++nt) {
        int col = nt * 16 + l15;
        int kt  = bkt[col];
#pragma unroll
        for (int r = 0; r < 8; ++r) {
            int rowc = m0 + r + half * 8;
            float dv = accs[nt][r] * SCALE_;
            if (bqt[rowc] == kt) dv = MASKV_;
            ps[rowc * 128 + col] = expf(dv - lrow[rowc]) * wrow[rowc];
        }
    }
    __syncthreads();

    // PV: out_tile += probs(16x128) @ V(128x16); k-outer, A loaded once per k
    {
        v8f z = {};
#pragma unroll
        for (int nt = 0; nt < 8; ++nt) accs[nt] = z;
    }
    {
        const float* prow = &ps[(m0 + l15) * 128];
#pragma unroll
        for (int k0 = 0; k0 < 128; k0 += 4) {
            v2f a;
            a.x = prow[k0 + kb];
            a.y = prow[k0 + kb + 1];
#pragma unroll
            for (int nt = 0; nt < 8; ++nt) {
                v2f bb;
                bb.x = vs[(k0 + kb)     * 128 + nt * 16 + l15];
                bb.y = vs[(k0 + kb + 1) * 128 + nt * 16 + l15];
                accs[nt] = __builtin_amdgcn_wmma_f32_16x16x4_f32(false, a, false, bb,
                                                                 (short)0, accs[nt], false, false);
            }
        }
    }
#pragma unroll
    for (int nt = 0; nt < 8; ++nt) {
#pragma unroll
        for (int r = 0; r < 8; ++r) {
            int rowc = m0 + r + half * 8;
            atomicAdd(&out[((size_t)(b * S_) + bqt[rowc]) * D_ + nt * 16 + l15], accs[nt][r]);
        }
    }
}

// ---------------------------------------------------------------------------
extern "C" void kernel_launch(void* const* d_in, const int* in_sizes, int n_in,
                              void* d_out, int out_size, void* d_ws, size_t ws_size,
                              hipStream_t stream) {
    (void)in_sizes; (void)n_in; (void)out_size; (void)ws_size;
    const float* qk   = (const float*)d_in[0];
    const float* v    = (const float*)d_in[1];
    const float* rotR = (const float*)d_in[2];

    float* out         = (float*)d_out;
    float* out_buckets = out + (size_t)B_ * S_ * D_;

    const size_t N1 = (size_t)B_ * H_ * S_;
    int*   bucket_i = (int*)d_ws;
    int*   sorted_t = bucket_i + N1;
    int*   posm     = sorted_t + N1;
    float* lse      = (float*)(posm + N1);
    float* wgt      = lse + N1;

    zero_kernel    <<<(B_ * S_ * D_ + 255) / 256, 256, 0, stream>>>(out, B_ * S_ * D_);
    hash_kernel    <<<B_ * (S_ / 16),             256, 0, stream>>>(qk, rotR, bucket_i, out_buckets);
    sort_kernel    <<<B_ * H_,                    256, 0, stream>>>(bucket_i, sorted_t, posm);
    attn_lse_kernel<<<B_ * C_,                    128, 0, stream>>>(qk, sorted_t, lse);
    combine_kernel <<<(B_ * S_ + 255) / 256,      256, 0, stream>>>(posm, lse, wgt);
    attn_out_kernel<<<B_ * C_,                    128, 0, stream>>>(qk, v, sorted_t, lse, wgt, out);
}